// UpSampleBlock_661424964110
// MI455X (gfx1250) — compile-verified
//
#include <hip/hip_runtime.h>

// ---------------------------------------------------------------------------
// Sparse upsample block (gather -> GEMM -> scatter) x3 + BN + ReLU, MI455X.
// Memory-bound (AI ~10 flop/B): f32 end-to-end, V_WMMA_F32_16X16X4_F32 for the
// per-offset GEMMs. Weights staged to LDS (b128), then cached per-wave in
// VGPRs and reused across a multi-tile grid-stride loop; scatter via native
// f32 global atomics; BN pipelines fully float4-vectorized.
// ---------------------------------------------------------------------------

typedef float v2f __attribute__((ext_vector_type(2)));
typedef float v8f __attribute__((ext_vector_type(8)));

static constexpr int N_OUT = 400000;
static constexpr int KOFF  = 27;
static constexpr int M_UP  = 100000;   // divisible by 16
static constexpr int M_C   = 120000;   // divisible by 16
static constexpr int COUT  = 32;

__device__ __forceinline__ void atomic_add_f32(float* p, float v) {
  __hip_atomic_fetch_add(p, v, __ATOMIC_RELAXED, __HIP_MEMORY_SCOPE_AGENT);
}

__device__ __forceinline__ float relu_f(float v) { return v > 0.f ? v : 0.f; }

// -------------------------------- utility ----------------------------------

__global__ __launch_bounds__(256) void zero_f4(float4* __restrict__ p, int n4) {
  int i = blockIdx.x * blockDim.x + threadIdx.x;
  const int stride = gridDim.x * blockDim.x;
  const float4 z = make_float4(0.f, 0.f, 0.f, 0.f);
  for (; i < n4; i += stride) p[i] = z;
}

__global__ __launch_bounds__(128) void zero_stats(float* __restrict__ stats) {
  stats[threadIdx.x] = 0.f;   // 128 floats: sum[32], sumsq[32], scale[32], shift[32]
}

// ---------------------- gather - WMMA GEMM - scatter -----------------------
// Block = 8 waves. Weights for kernel-offset k staged LDS-side with b128,
// then each wave caches its B stripe (2 x v2f per K-chunk) in VGPRs and
// grid-strides over 16-row M tiles. A (16x4 f32): lane = 16*half + m holds
// x[row][4c+2*half .. +1] (one aligned b64 gather per chunk). D layout
// (lane l, vgpr v -> M = v + 8*(l>=16), N = l%16) feeds the atomic scatter.
// All WMMA sit under wave-uniform control flow => EXEC all-ones.

template<int CIN>
__global__ __launch_bounds__(256)
void sparse_conv_wmma(const float* __restrict__ x,      // [*, CIN] gather source
                      const float* __restrict__ w,      // [K][CIN][COUT]
                      const int*   __restrict__ in_idx, // [K][M]
                      const int*   __restrict__ out_idx,// [K][M]
                      float*       __restrict__ out,    // [N_OUT][COUT]
                      int M)
{
  constexpr int NCH = CIN / 4;            // K-chunks of width 4
  __shared__ float wlds[CIN * COUT];

  const int k = blockIdx.y;
  {
    const float4* __restrict__ wk4 = (const float4*)(w + (size_t)k * (CIN * COUT));
    float4* wl4 = (float4*)wlds;
    for (int i = threadIdx.x; i < CIN * COUT / 4; i += 256) wl4[i] = wk4[i];
  }
  __syncthreads();

  const int lane = threadIdx.x & 31;
  const int l16  = lane & 15;
  const int half = lane >> 4;             // 0: K 0..1 / M 0..7 ; 1: K 2..3 / M 8..15

  // Cache this lane's B stripe in registers (CIN*COUT/32 = 2*CIN/4 v2f).
  v2f b0[NCH], b1[NCH];
#pragma unroll
  for (int c = 0; c < NCH; ++c) {
    const int kk = c * 4 + half * 2;
    b0[c].x = wlds[(kk    ) * COUT + l16];
    b0[c].y = wlds[(kk + 1) * COUT + l16];
    b1[c].x = wlds[(kk    ) * COUT + 16 + l16];
    b1[c].y = wlds[(kk + 1) * COUT + 16 + l16];
  }

  const int* __restrict__ inK  = in_idx  + (size_t)k * M;
  const int* __restrict__ outK = out_idx + (size_t)k * M;

  const int wave   = threadIdx.x >> 5;
  const int tiles  = M >> 4;
  const int stride = gridDim.x * 8;       // waves in grid.x

  for (int t = blockIdx.x * 8 + wave; t < tiles; t += stride) {
    const int m0 = t * 16;
    const int gr = inK[m0 + l16];
    const float* __restrict__ xr = x + (size_t)gr * CIN;

    v8f acc0 = {};                        // N = 0..15
    v8f acc1 = {};                        // N = 16..31
#pragma unroll
    for (int c = 0; c < NCH; ++c) {
      const int kk = c * 4 + half * 2;
      const v2f a = *(const v2f*)(xr + kk);   // 8B-aligned b64 gather
      acc0 = __builtin_amdgcn_wmma_f32_16x16x4_f32(false, a, false, b0[c],
                                                   (short)0, acc0, false, false);
      acc1 = __builtin_amdgcn_wmma_f32_16x16x4_f32(false, a, false, b1[c],
                                                   (short)0, acc1, false, false);
    }

#pragma unroll
    for (int v = 0; v < 8; ++v) {
      const int orow = outK[m0 + v + half * 8];
      float* dst = out + (size_t)orow * COUT;
      atomic_add_f32(dst + l16,      acc0[v]);
      atomic_add_f32(dst + 16 + l16, acc1[v]);
    }
  }
}

// ------------------------------- batch norm --------------------------------
// float4 streams; grid*block is a multiple of 8 float4s (= 32 channels), so
// each thread's 4-channel group is loop-invariant.

__global__ __launch_bounds__(256)
void bn_stats(const float4* __restrict__ h4, float* __restrict__ stats, int n) {
  int i = blockIdx.x * blockDim.x + threadIdx.x;
  const int stride = gridDim.x * blockDim.x;       // multiple of 8
  const int cb = (i * 4) & 31;
  float s0 = 0.f, s1 = 0.f, s2 = 0.f, s3 = 0.f;
  float q0 = 0.f, q1 = 0.f, q2 = 0.f, q3 = 0.f;
  const int total4 = n * 8;                        // n*32/4
  for (; i < total4; i += stride) {
    const float4 v = h4[i];
    s0 += v.x; q0 += v.x * v.x;
    s1 += v.y; q1 += v.y * v.y;
    s2 += v.z; q2 += v.z * v.z;
    s3 += v.w; q3 += v.w * v.w;
  }
  atomic_add_f32(&stats[cb + 0], s0);  atomic_add_f32(&stats[32 + cb + 0], q0);
  atomic_add_f32(&stats[cb + 1], s1);  atomic_add_f32(&stats[32 + cb + 1], q1);
  atomic_add_f32(&stats[cb + 2], s2);  atomic_add_f32(&stats[32 + cb + 2], q2);
  atomic_add_f32(&stats[cb + 3], s3);  atomic_add_f32(&stats[32 + cb + 3], q3);
}

__global__ __launch_bounds__(32)
void bn_finalize(float* __restrict__ stats,
                 const float* __restrict__ gamma,
                 const float* __restrict__ beta, int n) {
  const int c = threadIdx.x;
  const float inv_n = 1.f / (float)n;
  const float mean  = stats[c] * inv_n;
  const float var   = stats[32 + c] * inv_n - mean * mean;
  const float sc    = gamma[c] * rsqrtf(var + 1e-5f);
  stats[64 + c] = sc;
  stats[96 + c] = beta[c] - mean * sc;
}

// BN+ReLU into low half of concat rows, skip connection into high half.
__global__ __launch_bounds__(256)
void bn_apply_cat(const float4* __restrict__ h4,
                  const float4* __restrict__ skip4,
                  const float* __restrict__ stats,
                  float4* __restrict__ hcat4, int n) {
  int i = blockIdx.x * blockDim.x + threadIdx.x;
  const int stride = gridDim.x * blockDim.x;       // multiple of 8
  const int cb = (i * 4) & 31;
  const float4 sc = *(const float4*)(stats + 64 + cb);
  const float4 sh = *(const float4*)(stats + 96 + cb);
  const int total4 = n * 8;
  for (; i < total4; i += stride) {
    const int r  = i >> 3;                         // voxel row
    const int c4 = i & 7;                          // float4 slot within 32 ch
    float4 v = h4[i];
    v.x = relu_f(v.x * sc.x + sh.x);
    v.y = relu_f(v.y * sc.y + sh.y);
    v.z = relu_f(v.z * sc.z + sh.z);
    v.w = relu_f(v.w * sc.w + sh.w);
    hcat4[(size_t)r * 16 + c4]     = v;            // row stride = 64 ch = 16 f4
    hcat4[(size_t)r * 16 + 8 + c4] = skip4[i];
  }
}

__global__ __launch_bounds__(256)
void bn_apply(const float4* __restrict__ h4,
              const float* __restrict__ stats,
              float4* __restrict__ dst, int n) {
  int i = blockIdx.x * blockDim.x + threadIdx.x;
  const int stride = gridDim.x * blockDim.x;       // multiple of 8
  const int cb = (i * 4) & 31;
  const float4 sc = *(const float4*)(stats + 64 + cb);
  const float4 sh = *(const float4*)(stats + 96 + cb);
  const int total4 = n * 8;
  for (; i < total4; i += stride) {
    float4 v = h4[i];
    v.x = relu_f(v.x * sc.x + sh.x);
    v.y = relu_f(v.y * sc.y + sh.y);
    v.z = relu_f(v.z * sc.z + sh.z);
    v.w = relu_f(v.w * sc.w + sh.w);
    dst[i] = v;
  }
}

// --------------------------------- driver ----------------------------------

extern "C" void kernel_launch(void* const* d_in, const int* in_sizes, int n_in,
                              void* d_out, int out_size, void* d_ws, size_t ws_size,
                              hipStream_t stream) {
  (void)in_sizes; (void)n_in; (void)out_size; (void)ws_size;

  const float* x      = (const float*)d_in[0];
  const float* x_skip = (const float*)d_in[1];
  const float* w_up   = (const float*)d_in[2];
  // d_in[3] = b_up: cancels analytically under train-mode BN (mean shift).
  const float* g_up   = (const float*)d_in[4];
  const float* bt_up  = (const float*)d_in[5];
  const float* w1     = (const float*)d_in[6];
  const float* g1     = (const float*)d_in[7];
  const float* bt1    = (const float*)d_in[8];
  const float* w2     = (const float*)d_in[9];
  const float* g2     = (const float*)d_in[10];
  const float* bt2    = (const float*)d_in[11];
  const int* up_in    = (const int*)d_in[12];
  const int* up_out   = (const int*)d_in[13];
  const int* c1_in    = (const int*)d_in[14];
  const int* c1_out   = (const int*)d_in[15];
  const int* c2_in    = (const int*)d_in[16];
  const int* c2_out   = (const int*)d_in[17];

  float* out = (float*)d_out;

  float* wsf   = (float*)d_ws;
  float* h1    = wsf;                      // [N_OUT][32]  (reused as bn(conv1))
  float* hcat  = h1 + (size_t)N_OUT * 32;  // [N_OUT][64]  (reused as conv2 acc)
  float* h2    = hcat + (size_t)N_OUT * 64;// [N_OUT][32]
  float* stats = h2 + (size_t)N_OUT * 32;  // 128 floats (16B aligned)
  float* h3    = hcat;                     // alias: hcat free after conv1 gather

  const dim3 blk(256);
  const dim3 gz(2048);
  const dim3 g_ew(2048);
  // 8 waves/block, ~4 tiles/wave to amortize the register-cached B stripe.
  const dim3 g_up_conv((M_UP / 16 + 31) / 32, KOFF);
  const dim3 g_c_conv((M_C / 16 + 31) / 32, KOFF);
  const dim3 g_stats(512);                 // 512*256 threads, multiple of 8 f4

  // ---- stage 1: transposed sparse upconv onto skip grid, BN, ReLU --------
  zero_f4<<<gz, blk, 0, stream>>>((float4*)h1, N_OUT * 32 / 4);
  zero_stats<<<1, 128, 0, stream>>>(stats);
  sparse_conv_wmma<64><<<g_up_conv, blk, 0, stream>>>(x, w_up, up_in, up_out, h1, M_UP);
  bn_stats<<<g_stats, blk, 0, stream>>>((const float4*)h1, stats, N_OUT);
  bn_finalize<<<1, 32, 0, stream>>>(stats, g_up, bt_up, N_OUT);
  bn_apply_cat<<<g_ew, blk, 0, stream>>>((const float4*)h1, (const float4*)x_skip,
                                         stats, (float4*)hcat, N_OUT);

  // ---- stage 2: conv1 (2C -> C), BN, ReLU --------------------------------
  zero_f4<<<gz, blk, 0, stream>>>((float4*)h2, N_OUT * 32 / 4);
  zero_stats<<<1, 128, 0, stream>>>(stats);
  sparse_conv_wmma<64><<<g_c_conv, blk, 0, stream>>>(hcat, w1, c1_in, c1_out, h2, M_C);
  bn_stats<<<g_stats, blk, 0, stream>>>((const float4*)h2, stats, N_OUT);
  bn_finalize<<<1, 32, 0, stream>>>(stats, g1, bt1, N_OUT);
  bn_apply<<<g_ew, blk, 0, stream>>>((const float4*)h2, stats, (float4*)h1, N_OUT);

  // ---- stage 3: conv2 (C -> C), BN, ReLU ---------------------------------
  zero_f4<<<gz, blk, 0, stream>>>((float4*)h3, N_OUT * 32 / 4);
  zero_stats<<<1, 128, 0, stream>>>(stats);
  sparse_conv_wmma<32><<<g_c_conv, blk, 0, stream>>>(h1, w2, c2_in, c2_out, h3, M_C);
  bn_stats<<<g_stats, blk, 0, stream>>>((const float4*)h3, stats, N_OUT);
  bn_finalize<<<1, 32, 0, stream>>>(stats, g2, bt2, N_OUT);
  bn_apply<<<g_ew, blk, 0, stream>>>((const float4*)h3, stats, (float4*)out, N_OUT);
}